// AdaptiveAveragePooling_64218351010317
// MI455X (gfx1250) — compile-verified
//
#include <hip/hip_runtime.h>

typedef float v2f __attribute__((ext_vector_type(2)));
typedef float v4f __attribute__((ext_vector_type(4)));
typedef float v8f __attribute__((ext_vector_type(8)));

#define BATCH 8
#define H 224
#define W 224
#define C 256
#define OH 7
#define OW 7
#define WIN 32            // 224 / 7: exact disjoint windows
#define S 8               // row-chunks per window
#define ROWS_PER_CHUNK 4  // WIN / S
#define NWIN (BATCH * OH * OW)

// ---------------------------------------------------------------------------
// Kernel 1: streaming partial sums. One block per (window, row-chunk).
// grid.x = NWIN * S, block = 256 threads (8 wave32).
// Thread t: channel-group cg = t&63 (float4), sub-row = t>>6.
// Each thread: 32 contiguous nontemporal b128 loads (512B coalesced per wave).
// ---------------------------------------------------------------------------
__global__ __launch_bounds__(256) void pool_partial(const float* __restrict__ x,
                                                    float* __restrict__ ws) {
  const int blk = blockIdx.x;
  const int s   = blk & (S - 1);
  const int win = blk / S;
  const int ow  = win % OW;
  const int oh  = (win / OW) % OH;
  const int b   = win / (OW * OH);

  const int t   = threadIdx.x;
  const int cg  = t & 63;   // float4 channel group: channels 4*cg .. 4*cg+3
  const int sub = t >> 6;   // 0..3 -> row within this chunk

  const int h  = oh * WIN + s * ROWS_PER_CHUNK + sub;
  const int w0 = ow * WIN;

  const v4f* __restrict__ p =
      (const v4f*)x + ((((size_t)b * H + h) * W + w0) * C >> 2) + cg;

  v4f acc = {};
#pragma unroll
  for (int w = 0; w < WIN; ++w) {
    acc += __builtin_nontemporal_load(p + (size_t)w * (C / 4));
  }

  __shared__ v4f red[256];
  red[t] = acc;
  __syncthreads();
  if (t < 64) {
    v4f tot = red[t] + red[t + 64] + red[t + 128] + red[t + 192];
    ((v4f*)ws)[(size_t)blk * 64 + t] = tot;  // ws[blk][c], c = 4*t..4*t+3
  }
}

// ---------------------------------------------------------------------------
// Kernel 2: combine the S=8 chunk-partials per window via the CDNA5 matrix
// path: D = A(16x4, all 1/1024) x B(4x16 chunk-partials) + C, chained twice.
// Every row of D holds (1/1024) * sum_k B[k][n]; lane n (<16), VGPR0 of D is
// the finished output for channel n (per ISA 32-bit C/D layout).
// grid.x = NWIN, block = 256 (8 waves); each wave does 2 tiles of 16 channels.
// ---------------------------------------------------------------------------
__global__ __launch_bounds__(256) void pool_combine(const float* __restrict__ ws,
                                                    float* __restrict__ out) {
  const int win  = blockIdx.x;
  const int lane = threadIdx.x & 31;
  const int wave = threadIdx.x >> 5;

  const float scale = 1.0f / (float)(WIN * WIN);  // 1/1024
  v2f a;
  a.x = scale;
  a.y = scale;  // uniform: every element of the 16x4 A matrix is 1/1024

  const float* __restrict__ base = ws + (size_t)win * (S * C);
  const int krow = lane >> 4;  // B row striping: VGPR0 = K{0,1}, VGPR1 = K{2,3}

#pragma unroll
  for (int i = 0; i < 2; ++i) {
    const int ctile = wave + i * 8;           // 0..15
    const int c     = ctile * 16 + (lane & 15);

    v2f b0, b1;
    b0.x = base[(0 + krow) * C + c];  // chunks 0/1
    b0.y = base[(2 + krow) * C + c];  // chunks 2/3
    b1.x = base[(4 + krow) * C + c];  // chunks 4/5
    b1.y = base[(6 + krow) * C + c];  // chunks 6/7

    v8f acc = {};
    acc = __builtin_amdgcn_wmma_f32_16x16x4_f32(false, a, false, b0,
                                                (short)0, acc, false, false);
    acc = __builtin_amdgcn_wmma_f32_16x16x4_f32(false, a, false, b1,
                                                (short)0, acc, false, false);

    if (lane < 16) out[(size_t)win * C + c] = acc[0];
  }
}

extern "C" void kernel_launch(void* const* d_in, const int* in_sizes, int n_in,
                              void* d_out, int out_size, void* d_ws, size_t ws_size,
                              hipStream_t stream) {
  const float* x = (const float*)d_in[0];
  float* out = (float*)d_out;
  float* ws = (float*)d_ws;  // needs NWIN*S*C*4 = 3,211,264 bytes

  pool_partial<<<NWIN * S, 256, 0, stream>>>(x, ws);
  pool_combine<<<NWIN, 256, 0, stream>>>(ws, out);
}